// SpatialLocalAttention_35948876267796
// MI455X (gfx1250) — compile-verified
//
#include <hip/hip_runtime.h>
#include <cstddef>

// ---------------------------------------------------------------------------
// Shapes (fixed by the reference)
// ---------------------------------------------------------------------------
constexpr int Bsz   = 2;
constexpr int Lsz   = 4096;
constexpr int Dsz   = 256;
constexpr int Ksz   = 16;     // top-k neighbors
constexpr int Gsz   = 4;      // global latents
constexpr int Hsz   = 8;
constexpr int DHsz  = 64;
constexpr int PEsz  = 32;
constexpr int INNER = Hsz * DHsz;        // 512
constexpr int CSL   = 1 + Ksz + Gsz;     // 21 context slots
constexpr int NROW  = Bsz * Lsz;         // 8192
constexpr float SCALE = 0.125f;          // DH^-0.5

typedef float v2f __attribute__((ext_vector_type(2)));
typedef float v8f __attribute__((ext_vector_type(8)));

// ---------------------------------------------------------------------------
// fp32 WMMA GEMM  C[M,N] = A[M,K] @ B[K,N] (+ bias)
//   grid = (N/64, M/128), block = 128 (4 waves).
//   Each wave owns a 32x64 tile: 2 M-subtiles x 4 N-subtiles = 8 accumulators,
//   so 8 WMMAs amortize each group of 8 B-fragment loads (2x better than 4).
//   Fragments are register double-buffered: loads for k+4 are issued before
//   the WMMA group for k, so s_wait_loadcnt overlaps with the matrix pipe
//   instead of serializing (previous version waited to 0 before every WMMA).
//   V_WMMA_F32_16X16X4_F32 layout: A frag = 2 f32/lane (lane&15 = M row,
//   lane>>4 selects K pair {0,1}/{2,3}); B frag = 2 f32/lane (same K pair,
//   lane&15 = N col); C/D = 8 f32/lane (row r + 8*(lane>>4), col lane&15).
// ---------------------------------------------------------------------------
__global__ __launch_bounds__(128)
void gemm_wmma_f32_kernel(const float* __restrict__ A, int lda,
                          const float* __restrict__ B, int ldb,
                          float* __restrict__ C, int ldc,
                          int K, const float* __restrict__ bias)
{
    const int lane = threadIdx.x & 31;
    const int wave = threadIdx.x >> 5;
    const int row0 = blockIdx.y * 128 + wave * 32;   // 32 rows per wave
    const int col0 = blockIdx.x * 64;                // 64 cols per wave
    const int m16  = lane & 15;
    const int kOff = (lane >> 4) << 1;               // 0 or 2

    const float* Ap0 = A + (size_t)(row0 + m16) * lda + kOff;
    const float* Ap1 = Ap0 + (size_t)16 * lda;
    const float* Bp0 = B + (size_t)kOff * ldb + col0 + m16;

    v8f acc[2][4];
    const v8f zero = {0.f,0.f,0.f,0.f,0.f,0.f,0.f,0.f};
#pragma unroll
    for (int m = 0; m < 2; ++m)
#pragma unroll
        for (int t = 0; t < 4; ++t) acc[m][t] = zero;

    // prologue: fragments for k = 0
    v2f a[2], bf[4];
    a[0] = *(const v2f*)(Ap0);
    a[1] = *(const v2f*)(Ap1);
#pragma unroll
    for (int t = 0; t < 4; ++t)
        bf[t] = v2f{ Bp0[t * 16], Bp0[ldb + t * 16] };

    for (int k = 4; k < K; k += 4) {
        // prefetch fragments for this k (consumed next trip)
        v2f an[2], bn[4];
        an[0] = *(const v2f*)(Ap0 + k);
        an[1] = *(const v2f*)(Ap1 + k);
        const float* Bp = Bp0 + (size_t)k * ldb;
#pragma unroll
        for (int t = 0; t < 4; ++t)
            bn[t] = v2f{ Bp[t * 16], Bp[ldb + t * 16] };

        // WMMA group on previous iteration's fragments
#pragma unroll
        for (int t = 0; t < 4; ++t) {
            acc[0][t] = __builtin_amdgcn_wmma_f32_16x16x4_f32(
                false, a[0], false, bf[t], (short)0, acc[0][t], false, false);
            acc[1][t] = __builtin_amdgcn_wmma_f32_16x16x4_f32(
                false, a[1], false, bf[t], (short)0, acc[1][t], false, false);
        }
        a[0] = an[0]; a[1] = an[1];
#pragma unroll
        for (int t = 0; t < 4; ++t) bf[t] = bn[t];
    }

    // epilogue: last k-step
#pragma unroll
    for (int t = 0; t < 4; ++t) {
        acc[0][t] = __builtin_amdgcn_wmma_f32_16x16x4_f32(
            false, a[0], false, bf[t], (short)0, acc[0][t], false, false);
        acc[1][t] = __builtin_amdgcn_wmma_f32_16x16x4_f32(
            false, a[1], false, bf[t], (short)0, acc[1][t], false, false);
    }

    const int rHalf = (lane >> 4) << 3;
#pragma unroll
    for (int m = 0; m < 2; ++m) {
        const int rBase = row0 + m * 16 + rHalf;
#pragma unroll
        for (int t = 0; t < 4; ++t) {
            const int col = col0 + t * 16 + m16;
            const float bb = bias ? bias[col] : 0.f;
#pragma unroll
            for (int r = 0; r < 8; ++r)
                C[(size_t)(rBase + r) * ldc + col] = acc[m][t][r] + bb;
        }
    }
}

// ---------------------------------------------------------------------------
// K/V for the 8 global-latent rows (no RPE; zero-padded in ref).
// ---------------------------------------------------------------------------
__global__ __launch_bounds__(256)
void global_kv_kernel(const float* __restrict__ gl,
                      const float* __restrict__ Wk,
                      const float* __restrict__ Wv,
                      float* __restrict__ Kg, float* __restrict__ Vg)
{
    const int r   = blockIdx.x;          // 0..7 = b*4+g
    const int tid = threadIdx.x;
    __shared__ float sg[Dsz];
    sg[tid] = gl[r * Dsz + tid];
    __syncthreads();
    for (int f = tid; f < INNER; f += 256) {
        float aK = 0.f, aV = 0.f;
        for (int d = 0; d < Dsz; ++d) {
            const float x = sg[d];
            aK += x * Wk[(size_t)d * INNER + f];
            aV += x * Wv[(size_t)d * INNER + f];
        }
        Kg[r * INNER + f] = aK;
        Vg[r * INNER + f] = aV;
    }
}

// ---------------------------------------------------------------------------
// Fused attention. One block (256 thr = 8 waves, wave h = head h) per query
// (b,l). K tile built in LDS (base gather + 32-wide RPE GEMV with weight
// columns cached in registers), per-head wave32 softmax over 21 slots, then
// the same LDS buffer is reused for the V tile (keeps static LDS ~48 KB).
// Row stride 517 floats -> lane j hits bank (5j+d) mod 64: conflict-free.
// ---------------------------------------------------------------------------
constexpr int KST = 517;

__global__ __launch_bounds__(256)
void attn_kernel(const float* __restrict__ Qb, const float* __restrict__ Kb,
                 const float* __restrict__ Vb, const float* __restrict__ Kg,
                 const float* __restrict__ Vg, const int* __restrict__ topk,
                 const float* __restrict__ rpe, const float* __restrict__ srpe,
                 const float* __restrict__ dist,
                 const float* __restrict__ Wk, const float* __restrict__ Wv,
                 const float* __restrict__ log_sigma,
                 const float* __restrict__ gbias,
                 float* __restrict__ AO)
{
    __shared__ float sQ[INNER];
    __shared__ float sKV[CSL * KST];     // K tile, then reused for V tile
    __shared__ float sR[(1 + Ksz) * PEsz];
    __shared__ int   sIdx[Ksz];
    __shared__ float sDist[Ksz];

    const int tid = threadIdx.x;
    const int bid = blockIdx.x;          // b*L + l
    const int b   = bid >> 12;
    const int l   = bid & (Lsz - 1);

    if (tid < Ksz) {
        sIdx[tid]  = topk[bid * Ksz + tid];
        sDist[tid] = dist[bid * Ksz + tid];
    }
    for (int i = tid; i < (1 + Ksz) * PEsz; i += 256) {
        const int j = i >> 5, p = i & 31;
        sR[i] = (j == 0) ? srpe[(size_t)bid * PEsz + p]
                         : rpe[((size_t)bid * Ksz + (j - 1)) * PEsz + p];
    }
    for (int i = tid; i < INNER; i += 256)
        sQ[i] = Qb[(size_t)bid * INNER + i];
    __syncthreads();

    // ---- build K tile --------------------------------------------------
    {
        const float* Wr = Wk + (size_t)Dsz * INNER;   // RPE rows of Wk
        for (int f = tid; f < INNER; f += 256) {
            float w[PEsz];
#pragma unroll
            for (int p = 0; p < PEsz; ++p) w[p] = Wr[(size_t)p * INNER + f];
#pragma unroll 1
            for (int j = 0; j < 1 + Ksz; ++j) {
                const int src = (j == 0) ? l : sIdx[j - 1];
                float acc = Kb[((size_t)(b * Lsz + src)) * INNER + f];
#pragma unroll
                for (int p = 0; p < PEsz; ++p) acc += sR[j * PEsz + p] * w[p];
                sKV[j * KST + f] = acc;
            }
#pragma unroll
            for (int g = 0; g < Gsz; ++g)
                sKV[(1 + Ksz + g) * KST + f] = Kg[(b * Gsz + g) * INNER + f];
        }
    }
    __syncthreads();

    // ---- scores + softmax (wave h handles head h; lane j owns slot j) --
    const int h = tid >> 5, lane = tid & 31;
    float wgt;
    {
        float score = -3.0e38f;
        if (lane < CSL) {
            float s = 0.f;
            const int kb = lane * KST + h * DHsz;
#pragma unroll 8
            for (int d = 0; d < DHsz; ++d) s += sQ[h * DHsz + d] * sKV[kb + d];
            s *= SCALE;
            float biasv;
            if (lane == 0) {
                biasv = 0.f;                                  // self: dist 0
            } else if (lane < 1 + Ksz) {
                const float dd  = sDist[lane - 1];
                const float ssq = expf(2.f * log_sigma[h]);   // sigma^2
                biasv = -(dd * dd) * (0.5f / ssq);
            } else {
                biasv = gbias[0];
            }
            score = s + biasv;
        }
        float m = score;
        for (int off = 16; off > 0; off >>= 1)
            m = fmaxf(m, __shfl_xor(m, off, 32));
        const float e = (lane < CSL) ? expf(score - m) : 0.f;
        float sum = e;
        for (int off = 16; off > 0; off >>= 1)
            sum += __shfl_xor(sum, off, 32);
        wgt = e / sum;
    }
    __syncthreads();   // all waves done reading K tile before overwrite

    // ---- build V tile (same LDS buffer) --------------------------------
    {
        const float* Wr = Wv + (size_t)Dsz * INNER;
        for (int f = tid; f < INNER; f += 256) {
            float w[PEsz];
#pragma unroll
            for (int p = 0; p < PEsz; ++p) w[p] = Wr[(size_t)p * INNER + f];
#pragma unroll 1
            for (int j = 0; j < 1 + Ksz; ++j) {
                const int src = (j == 0) ? l : sIdx[j - 1];
                float acc = Vb[((size_t)(b * Lsz + src)) * INNER + f];
#pragma unroll
                for (int p = 0; p < PEsz; ++p) acc += sR[j * PEsz + p] * w[p];
                sKV[j * KST + f] = acc;
            }
#pragma unroll
            for (int g = 0; g < Gsz; ++g)
                sKV[(1 + Ksz + g) * KST + f] = Vg[(b * Gsz + g) * INNER + f];
        }
    }
    __syncthreads();

    // ---- weighted sum of V: each lane owns dims (lane, lane+32) --------
    float a0 = 0.f, a1 = 0.f;
    const int vb = h * DHsz + lane;
#pragma unroll 1
    for (int j = 0; j < CSL; ++j) {
        const float wj = __shfl(wgt, j, 32);
        a0 += wj * sKV[j * KST + vb];
        a1 += wj * sKV[j * KST + vb + 32];
    }
    AO[(size_t)bid * INNER + h * DHsz + lane]      = a0;
    AO[(size_t)bid * INNER + h * DHsz + lane + 32] = a1;
}

// ---------------------------------------------------------------------------
// Launch
// ---------------------------------------------------------------------------
extern "C" void kernel_launch(void* const* d_in, const int* in_sizes, int n_in,
                              void* d_out, int out_size, void* d_ws, size_t ws_size,
                              hipStream_t stream)
{
    const float* spatial   = (const float*)d_in[0];
    const int*   topk      = (const int*)  d_in[1];
    const float* rpe       = (const float*)d_in[2];
    const float* self_rpe  = (const float*)d_in[3];
    const float* distances = (const float*)d_in[4];
    const float* glat      = (const float*)d_in[5];
    const float* Wq        = (const float*)d_in[6];
    const float* Wk        = (const float*)d_in[7];
    const float* Wv        = (const float*)d_in[8];
    const float* Wo        = (const float*)d_in[9];
    const float* bo        = (const float*)d_in[10];
    const float* log_sigma = (const float*)d_in[11];
    const float* gbias     = (const float*)d_in[12];
    float* out = (float*)d_out;
    float* ws  = (float*)d_ws;

    const size_t rowsz = (size_t)NROW * INNER;   // 8192*512
    float* Qb = ws;
    float* Kb = ws + rowsz;
    float* Vb = ws + 2 * rowsz;
    float* AO = ws + 3 * rowsz;
    float* Kg = ws + 4 * rowsz;
    float* Vg = Kg + Bsz * Gsz * INNER;

    dim3 blk(128);
    // base projections: [8192,256] @ [256,512] (Wk/Wv: top 256 rows only)
    gemm_wmma_f32_kernel<<<dim3(INNER / 64, NROW / 128), blk, 0, stream>>>(
        spatial, Dsz, Wq, INNER, Qb, INNER, Dsz, nullptr);
    gemm_wmma_f32_kernel<<<dim3(INNER / 64, NROW / 128), blk, 0, stream>>>(
        spatial, Dsz, Wk, INNER, Kb, INNER, Dsz, nullptr);
    gemm_wmma_f32_kernel<<<dim3(INNER / 64, NROW / 128), blk, 0, stream>>>(
        spatial, Dsz, Wv, INNER, Vb, INNER, Dsz, nullptr);

    global_kv_kernel<<<Bsz * Gsz, 256, 0, stream>>>(glat, Wk, Wv, Kg, Vg);

    attn_kernel<<<NROW, 256, 0, stream>>>(Qb, Kb, Vb, Kg, Vg, topk, rpe,
                                          self_rpe, distances, Wk, Wv,
                                          log_sigma, gbias, AO);

    // output projection: [8192,512] @ [512,256] + bo
    gemm_wmma_f32_kernel<<<dim3(Dsz / 64, NROW / 128), blk, 0, stream>>>(
        AO, INNER, Wo, Dsz, out, Dsz, INNER, bo);
}